// GraphSAGE_46755013984826
// MI455X (gfx1250) — compile-verified
//
#include <hip/hip_runtime.h>
#include <hip/hip_bf16.h>

typedef __attribute__((ext_vector_type(2))) float v2f;
typedef __attribute__((ext_vector_type(8))) float v8f;

#define D 128
#define LDS_STRIDE 132   // 132 floats = 528 B = 33*16 -> rows stay 16B-aligned
#define ROWS 64          // rows per block = 4 sub-tiles of 16
#define NT 4

// ---------------------------------------------------------------------------
// 4x (16x16) output tiles per wave over K=128 with V_WMMA_F32_16X16X4_F32.
// A tiles in LDS (ROWS x 128, stride LDS_STRIDE). WT is the TRANSPOSED weight
// (WT[n][k] = W[k][n]) so each lane's B pair (ka, ka+1) is one b64 load.
// Fragment layouts (CDNA5 ISA 7.12.2):
//   A(m,k): vgpr=k&1, lane=m+16*(k>>1)
//   B(k,n): vgpr=k&1, lane=n+16*(k>>1)
//   C(m,n): vgpr=m&7, lane=n+16*(m>>3)
// ---------------------------------------------------------------------------
__device__ __forceinline__ void wmma_gemm_tiles(const float* Alds,
                                                const float* __restrict__ WT,
                                                int nb, int hi, int m,
                                                v8f acc[NT]) {
#pragma unroll
  for (int k0 = 0; k0 < D; k0 += 4) {
    const int ka = k0 + 2 * hi;                       // this lane's k pair
    const v2f b = *(const v2f*)&WT[(size_t)nb * D + ka];
#pragma unroll
    for (int t = 0; t < NT; ++t) {
      const v2f a = *(const v2f*)&Alds[(t * 16 + m) * LDS_STRIDE + ka];
      acc[t] = __builtin_amdgcn_wmma_f32_16x16x4_f32(false, a, false, b,
                                                     (short)0, acc[t], false, false);
    }
  }
}

// WT[n*128+k] = W[k*128+n]
__global__ void transpose128_kernel(const float* __restrict__ W,
                                    float* __restrict__ WT) {
  const int idx = blockIdx.x * 256 + threadIdx.x;     // grid covers 128*128
  if (idx < D * D) {
    const int k = idx >> 7, n = idx & (D - 1);
    WT[n * D + k] = W[idx];
  }
}

// out_raw = X @ W + b ; optionally out_relu = relu(out_raw)
__global__ void __launch_bounds__(256)
fc_gemm_kernel(const float* __restrict__ X, const float* __restrict__ WT,
               const float* __restrict__ bias, float* __restrict__ out_raw,
               float* __restrict__ out_relu, int n_rows) {
  __shared__ float Alds[ROWS * LDS_STRIDE];
  const int row0 = blockIdx.x * ROWS;
  const float4 z4 = {0.f, 0.f, 0.f, 0.f};
  for (int t = threadIdx.x; t < ROWS * (D / 4); t += 256) {
    const int r = t >> 5, c4 = t & 31;
    const int row = row0 + r;
    float4 v = z4;
    if (row < n_rows) v = *(const float4*)&X[(size_t)row * D + c4 * 4];
    *(float4*)&Alds[r * LDS_STRIDE + c4 * 4] = v;
  }
  __syncthreads();

  const int lane = threadIdx.x & 31;
  const int wv   = threadIdx.x >> 5;
  const int hi   = lane >> 4;
  const int m    = lane & 15;
  const int nb   = wv * 16 + m;                       // this lane's output col

  v8f acc[NT] = {};
  wmma_gemm_tiles(Alds, WT, nb, hi, m, acc);

  const float bn = bias[nb];
#pragma unroll
  for (int t = 0; t < NT; ++t) {
#pragma unroll
    for (int r = 0; r < 8; ++r) {
      const int row = row0 + t * 16 + r + 8 * hi;
      if (row < n_rows) {
        const float v = acc[t][r] + bn;
        out_raw[(size_t)row * D + nb] = v;
        if (out_relu) out_relu[(size_t)row * D + nb] = fmaxf(v, 0.0f);
      }
    }
  }
}

// out = Hdst @ Wself + (Msg * invdeg[row]) @ Wneigh + b ; optional relu
__global__ void __launch_bounds__(256)
sage_gemm_kernel(const float* __restrict__ Hdst, const float* __restrict__ Msg,
                 const float* __restrict__ invdeg,
                 const float* __restrict__ WselfT, const float* __restrict__ WneighT,
                 const float* __restrict__ bias, float* __restrict__ out,
                 int n_rows, int do_relu) {
  __shared__ float Ad[ROWS * LDS_STRIDE];   // dst features
  __shared__ float An[ROWS * LDS_STRIDE];   // mean-neighbor features
  const int row0 = blockIdx.x * ROWS;
  const float4 z4 = {0.f, 0.f, 0.f, 0.f};
  for (int t = threadIdx.x; t < ROWS * (D / 4); t += 256) {
    const int r = t >> 5, c4 = t & 31;
    const int row = row0 + r;
    float4 vd = z4, vn = z4;
    if (row < n_rows) {
      vd = *(const float4*)&Hdst[(size_t)row * D + c4 * 4];
      const float s = invdeg[row];
      float4 mm = *(const float4*)&Msg[(size_t)row * D + c4 * 4];
      vn.x = mm.x * s; vn.y = mm.y * s; vn.z = mm.z * s; vn.w = mm.w * s;
    }
    *(float4*)&Ad[r * LDS_STRIDE + c4 * 4] = vd;
    *(float4*)&An[r * LDS_STRIDE + c4 * 4] = vn;
  }
  __syncthreads();

  const int lane = threadIdx.x & 31;
  const int wv   = threadIdx.x >> 5;
  const int hi   = lane >> 4;
  const int m    = lane & 15;
  const int nb   = wv * 16 + m;

  v8f acc[NT] = {};
  wmma_gemm_tiles(Ad, WselfT, nb, hi, m, acc);
  wmma_gemm_tiles(An, WneighT, nb, hi, m, acc);

  const float bn = bias[nb];
#pragma unroll
  for (int t = 0; t < NT; ++t) {
#pragma unroll
    for (int r = 0; r < 8; ++r) {
      const int row = row0 + t * 16 + r + 8 * hi;
      if (row < n_rows) {
        float v = acc[t][r] + bn;
        if (do_relu) v = fmaxf(v, 0.0f);
        out[(size_t)row * D + nb] = v;
      }
    }
  }
}

__global__ void zero_kernel(float* __restrict__ p, long n) {
  long i = (long)blockIdx.x * blockDim.x + threadIdx.x;
  const long stride = (long)gridDim.x * blockDim.x;
  for (; i < n; i += stride) p[i] = 0.0f;
}

__global__ void degree_kernel(const int* __restrict__ dst, int n_edges,
                              float* __restrict__ deg) {
  const int e = blockIdx.x * blockDim.x + threadIdx.x;
  if (e < n_edges) atomicAdd(&deg[dst[e]], 1.0f);
}

__global__ void invert_deg_kernel(float* __restrict__ deg, int n) {
  const int i = blockIdx.x * blockDim.x + threadIdx.x;
  if (i < n) deg[i] = 1.0f / fmaxf(deg[i], 1.0f);
}

// 32 lanes per edge; each lane atomically adds 4 of the 128 feature floats.
__global__ void __launch_bounds__(256)
scatter_add_kernel(const float* __restrict__ h, const int* __restrict__ src,
                   const int* __restrict__ dst, int n_edges,
                   float* __restrict__ msg) {
  const int e = (int)(((long)blockIdx.x * blockDim.x + threadIdx.x) >> 5);
  const int lane = threadIdx.x & 31;
  if (e >= n_edges) return;
  const float* s = h + (size_t)src[e] * D;
  float* d = msg + (size_t)dst[e] * D;
#pragma unroll
  for (int j = 0; j < 4; ++j)
    atomicAdd(&d[lane + 32 * j], s[lane + 32 * j]);
}

static inline int cdiv(long a, long b) { return (int)((a + b - 1) / b); }

extern "C" void kernel_launch(void* const* d_in, const int* in_sizes, int n_in,
                              void* d_out, int out_size, void* d_ws, size_t ws_size,
                              hipStream_t stream) {
  // ---- inputs (setup_inputs insertion order) ----
  const float* feat_s = (const float*)d_in[0];
  const float* feat_c = (const float*)d_in[1];
  const float* feat_l = (const float*)d_in[2];
  const int* us_src = (const int*)d_in[3];
  const int* us_dst = (const int*)d_in[4];
  const int* te_src = (const int*)d_in[5];
  const int* te_dst = (const int*)d_in[6];
  // params flattened in insertion order:
  // fc_student{W,b}, fc_concept{W,b}, fc_lecture{W,b},
  // layers[i]{understands{W_self,W_neigh,b}, teaches{W_self,W_neigh,b}}
  const float* fcS_W = (const float*)d_in[7];
  const float* fcS_b = (const float*)d_in[8];
  const float* fcC_W = (const float*)d_in[9];
  const float* fcC_b = (const float*)d_in[10];
  const float* fcL_W = (const float*)d_in[11];
  const float* fcL_b = (const float*)d_in[12];

  const int nS = in_sizes[0] / D;
  const int nC = in_sizes[1] / D;
  const int nL = in_sizes[2] / D;
  const int eUS = in_sizes[3];
  const int eTE = in_sizes[5];

  // ---- outputs: concat(h_s, h_c, h_l) ----
  float* out_s = (float*)d_out;
  float* out_c = out_s + (size_t)nS * D;
  float* out_l = out_c + (size_t)nC * D;

  // ---- workspace layout (floats) ----
  float* ws = (float*)d_ws;
  float* h_s_raw  = ws;                              // nS*D (layer-0 src, pre-relu)
  float* h_c0     = h_s_raw + (size_t)nS * D;        // nC*D
  float* h_c1     = h_c0 + (size_t)nC * D;           // nC*D
  float* h_l0     = h_c1 + (size_t)nC * D;           // nL*D
  float* h_l1     = h_l0 + (size_t)nL * D;           // nL*D
  float* msg_c    = h_l1 + (size_t)nL * D;           // nC*D
  float* msg_l    = msg_c + (size_t)nC * D;          // nL*D
  float* invdeg_c = msg_l + (size_t)nL * D;          // nC
  float* invdeg_l = invdeg_c + nC;                   // nL
  float* wt_base  = invdeg_l + nL;                   // 15 * 128*128 transposed W

  // ---- transpose all weight matrices once (b64 B-fragment loads later) ----
  // wt[0..2] = fcS, fcC, fcL ; wt[3+4i+{0,1,2,3}] = uWself,uWneigh,tWself,tWneigh
  const float* wsrc[15];
  wsrc[0] = fcS_W; wsrc[1] = fcC_W; wsrc[2] = fcL_W;
  for (int i = 0; i < 3; ++i) {
    wsrc[3 + 4 * i + 0] = (const float*)d_in[13 + 6 * i + 0];   // und W_self
    wsrc[3 + 4 * i + 1] = (const float*)d_in[13 + 6 * i + 1];   // und W_neigh
    wsrc[3 + 4 * i + 2] = (const float*)d_in[13 + 6 * i + 3];   // tea W_self
    wsrc[3 + 4 * i + 3] = (const float*)d_in[13 + 6 * i + 4];   // tea W_neigh
  }
  float* wt[15];
  for (int i = 0; i < 15; ++i) {
    wt[i] = wt_base + (size_t)i * D * D;
    transpose128_kernel<<<64, 256, 0, stream>>>(wsrc[i], wt[i]);
  }

  // ---- per-type input projections (WMMA GEMMs) ----
  // out_s doubles as the relu'd student embedding used as src for layers 1,2
  // and is the final student output (relu twice == relu once; students are a
  // passthrough in the last layer).
  fc_gemm_kernel<<<cdiv(nS, ROWS), 256, 0, stream>>>(feat_s, wt[0], fcS_b,
                                                     h_s_raw, out_s, nS);
  fc_gemm_kernel<<<cdiv(nC, ROWS), 256, 0, stream>>>(feat_c, wt[1], fcC_b,
                                                     h_c0, nullptr, nC);
  fc_gemm_kernel<<<cdiv(nL, ROWS), 256, 0, stream>>>(feat_l, wt[2], fcL_b,
                                                     h_l0, nullptr, nL);

  // ---- degrees (constant across layers) -> inverse ----
  zero_kernel<<<cdiv(nC, 256), 256, 0, stream>>>(invdeg_c, nC);
  zero_kernel<<<cdiv(nL, 256), 256, 0, stream>>>(invdeg_l, nL);
  degree_kernel<<<cdiv(eUS, 256), 256, 0, stream>>>(us_dst, eUS, invdeg_c);
  degree_kernel<<<cdiv(eTE, 256), 256, 0, stream>>>(te_dst, eTE, invdeg_l);
  invert_deg_kernel<<<cdiv(nC, 256), 256, 0, stream>>>(invdeg_c, nC);
  invert_deg_kernel<<<cdiv(nL, 256), 256, 0, stream>>>(invdeg_l, nL);

  // ---- 3 conv layers (ping-pong; last layer writes straight to d_out) ----
  const float* hs_src[3] = { h_s_raw, out_s, out_s };
  const float* hc_in[3]  = { h_c0, h_c1, h_c0 };
  float*       hc_out[3] = { h_c1, h_c0, out_c };
  const float* hl_in[3]  = { h_l0, h_l1, h_l0 };
  float*       hl_out[3] = { h_l1, h_l0, out_l };

  for (int i = 0; i < 3; ++i) {
    const float* u_b = (const float*)d_in[13 + 6 * i + 2];
    const float* t_b = (const float*)d_in[13 + 6 * i + 5];
    const int relu = (i != 2);

    // understands: student -> concept
    zero_kernel<<<cdiv((long)nC * D, 256L * 8), 256, 0, stream>>>(msg_c, (long)nC * D);
    scatter_add_kernel<<<cdiv((long)eUS * 32, 256), 256, 0, stream>>>(
        hs_src[i], us_src, us_dst, eUS, msg_c);
    // teaches: concept -> lecture (reads hc_in[i], not modified this layer)
    zero_kernel<<<cdiv((long)nL * D, 256L * 8), 256, 0, stream>>>(msg_l, (long)nL * D);
    scatter_add_kernel<<<cdiv((long)eTE * 32, 256), 256, 0, stream>>>(
        hc_in[i], te_src, te_dst, eTE, msg_l);

    sage_gemm_kernel<<<cdiv(nC, ROWS), 256, 0, stream>>>(
        hc_in[i], msg_c, invdeg_c, wt[3 + 4 * i + 0], wt[3 + 4 * i + 1],
        u_b, hc_out[i], nC, relu);
    sage_gemm_kernel<<<cdiv(nL, ROWS), 256, 0, stream>>>(
        hl_in[i], msg_l, invdeg_l, wt[3 + 4 * i + 2], wt[3 + 4 * i + 3],
        t_b, hl_out[i], nL, relu);
  }
}